// Codec_42520176230729
// MI455X (gfx1250) — compile-verified
//
#include <hip/hip_runtime.h>
#include <math.h>

#define TPB 256

static constexpr int    IMGS    = 96;
static constexpr size_t HW      = 512ull * 512;           // 262144
static constexpr size_t A_ELEMS = 96ull * 512 * 256;      // level-0 xe (compacted)
static constexpr size_t B_ELEMS = 96ull * 256 * 256;      // level-0 ye (compacted)
static constexpr int    HIST_N  = IMGS * 256;

// ---------- CDNA5 async global->LDS copy (gfx1250) ----------
#if defined(__gfx1250__) && __has_builtin(__builtin_amdgcn_global_load_async_to_lds_b128)
#define HAVE_ASYNC_LDS 1
#endif

typedef int v4i_t __attribute__((vector_size(16)));
typedef __attribute__((address_space(1))) v4i_t* gptr_v4i;
typedef __attribute__((address_space(3))) v4i_t* lptr_v4i;

__device__ __forceinline__ void async_copy_b128(void* lds_dst, const void* gsrc) {
#ifdef HAVE_ASYNC_LDS
  __builtin_amdgcn_global_load_async_to_lds_b128(
      (gptr_v4i)(v4i_t*)const_cast<void*>(gsrc),
      (lptr_v4i)(v4i_t*)lds_dst, 0, 0);
#else
  *(float4*)lds_dst = *(const float4*)gsrc;
#endif
}

__device__ __forceinline__ void wait_async_all() {
#if __has_builtin(__builtin_amdgcn_s_wait_asynccnt)
  __builtin_amdgcn_s_wait_asynccnt(0);
#elif defined(HAVE_ASYNC_LDS)
  asm volatile("s_wait_asynccnt 0x0" ::: "memory");
#endif
}

// ---------- histogram helper (binning identical for input and deltas) ----------
__device__ __forceinline__ void hist_add(unsigned* h, float d) {
  if (d >= -1.0f && d <= 1.0f) {
    int b = (int)floorf((d + 1.0f) * 128.0f);
    b = b < 0 ? 0 : (b > 255 ? 255 : b);
    atomicAdd(&h[b], 1u);
  }
}

// =====================================================================
// k0: zero stat buffers
// =====================================================================
__global__ void k0_init(unsigned* gh0, unsigned* gh1, float* gss) {
  int i = blockIdx.x * TPB + threadIdx.x;
  if (i < HIST_N) { gh0[i] = 0u; gh1[i] = 0u; }
  if (i < 2) gss[i] = 0.0f;
}

// =====================================================================
// k1: level-0 x-lift over rows. Input stats, xo stats, write compacted xe.
// grid = 96 images * 64 row-blocks (8 rows each), 256 threads.
// =====================================================================
__global__ __launch_bounds__(TPB) void k1_xlift_stats(
    const float* __restrict__ x,
    const float* __restrict__ px, const float* __restrict__ ux,
    const float* __restrict__ cx, const float* __restrict__ rx,
    float* __restrict__ A, unsigned* __restrict__ gh0,
    unsigned* __restrict__ gh1, float* __restrict__ gss)
{
  __shared__ __align__(16) float t[8 * 512];
  __shared__ unsigned h0[4 * 256];
  __shared__ unsigned h1[4 * 256];
  __shared__ float ssh[2];
  const int tid = threadIdx.x;
  const int img = blockIdx.x >> 6;
  const int rb  = blockIdx.x & 63;

  for (int i = tid; i < 1024; i += TPB) { h0[i] = 0u; h1[i] = 0u; }
  if (tid < 2) ssh[tid] = 0.0f;
  __syncthreads();

  unsigned* mh0 = &h0[((tid >> 5) & 3) * 256];
  unsigned* mh1 = &h1[((tid >> 5) & 3) * 256];

  const float p0 = px[0], p1 = px[1], p2 = px[2];
  const float u0 = ux[0], u1 = ux[1], u2 = ux[2];
  const float cc0 = cx[0], cc1 = cx[1], cc2 = cx[2];
  const float rr0 = rx[0], rr1 = rx[1], rr2 = rx[2];

  // ---- load 8x512 tile + input stats ----
  const float* src = x + (size_t)img * HW + (size_t)rb * (8 * 512);
  float ss0 = 0.0f;
  for (int q = tid; q < 1024; q += TPB) {
    float4 v = ((const float4*)src)[q];
    ((float4*)t)[q] = v;
    ss0 += v.x * v.x + v.y * v.y + v.z * v.z + v.w * v.w;
    hist_add(mh0, v.x); hist_add(mh0, v.y);
    hist_add(mh0, v.z); hist_add(mh0, v.w);
  }
  __syncthreads();

  // ---- in-place x lifting (evens = xe at 2j, odds = xo at 2j+1), M=256 ----
  // step1: odd -= px * evens
  for (int it = tid; it < 2048; it += TPB) {
    int r = it >> 8, j = it & 255;
    float em = (j > 0)   ? t[r * 512 + 2 * j - 2] : 0.0f;
    float e0 = t[r * 512 + 2 * j];
    float ep = (j < 255) ? t[r * 512 + 2 * j + 2] : 0.0f;
    t[r * 512 + 2 * j + 1] -= p0 * em + p1 * e0 + p2 * ep;
  }
  __syncthreads();
  // step2: even += ux * odds
  for (int it = tid; it < 2048; it += TPB) {
    int r = it >> 8, j = it & 255;
    float om = (j > 0)   ? t[r * 512 + 2 * j - 1] : 0.0f;
    float o0 = t[r * 512 + 2 * j + 1];
    float op = (j < 255) ? t[r * 512 + 2 * j + 3] : 0.0f;
    t[r * 512 + 2 * j] += u0 * om + u1 * o0 + u2 * op;
  }
  __syncthreads();
  // step3: odd -= cx * evens
  for (int it = tid; it < 2048; it += TPB) {
    int r = it >> 8, j = it & 255;
    float em = (j > 0)   ? t[r * 512 + 2 * j - 2] : 0.0f;
    float e0 = t[r * 512 + 2 * j];
    float ep = (j < 255) ? t[r * 512 + 2 * j + 2] : 0.0f;
    t[r * 512 + 2 * j + 1] -= cc0 * em + cc1 * e0 + cc2 * ep;
  }
  __syncthreads();
  // step4: even += rx * odds
  for (int it = tid; it < 2048; it += TPB) {
    int r = it >> 8, j = it & 255;
    float om = (j > 0)   ? t[r * 512 + 2 * j - 1] : 0.0f;
    float o0 = t[r * 512 + 2 * j + 1];
    float op = (j < 255) ? t[r * 512 + 2 * j + 3] : 0.0f;
    t[r * 512 + 2 * j] += rr0 * om + rr1 * o0 + rr2 * op;
  }
  __syncthreads();

  // ---- xo stats + write compacted xe ----
  float ss1 = 0.0f;
  for (int it = tid; it < 2048; it += TPB) {
    int r = it >> 8, j = it & 255;
    float v = t[r * 512 + 2 * j + 1];
    float d = fmodf(v + 1.0f, 2.0f) - 1.0f;
    ss1 += d * d;
    hist_add(mh1, d);
    A[((size_t)img * 512 + rb * 8 + r) * 256 + j] = t[r * 512 + 2 * j];
  }
  atomicAdd(&ssh[0], ss0);
  atomicAdd(&ssh[1], ss1);
  __syncthreads();
  for (int b = tid; b < 256; b += TPB) {
    unsigned s0 = h0[b] + h0[256 + b] + h0[512 + b] + h0[768 + b];
    unsigned s1 = h1[b] + h1[256 + b] + h1[512 + b] + h1[768 + b];
    if (s0) atomicAdd(&gh0[img * 256 + b], s0);
    if (s1) atomicAdd(&gh1[img * 256 + b], s1);
  }
  if (tid == 0) { atomicAdd(&gss[0], ssh[0]); atomicAdd(&gss[1], ssh[1]); }
}

// =====================================================================
// k2: level-0 y-lift over columns. 512x32 tile async-loaded into LDS.
// yo stats, write compacted ye. grid = 96 images * 8 col-tiles.
// =====================================================================
__global__ __launch_bounds__(TPB) void k2_ylift_stats(
    const float* __restrict__ A,
    const float* __restrict__ py, const float* __restrict__ uy,
    const float* __restrict__ cy, const float* __restrict__ ry,
    float* __restrict__ B, unsigned* __restrict__ gh1,
    float* __restrict__ gss)
{
  __shared__ __align__(16) float t[512 * 32];
  __shared__ unsigned h1[4 * 256];
  __shared__ float ssh;
  const int tid   = threadIdx.x;
  const int img   = blockIdx.x >> 3;
  const int cbase = (blockIdx.x & 7) * 32;

  for (int i = tid; i < 1024; i += TPB) h1[i] = 0u;
  if (tid == 0) ssh = 0.0f;

  // async load 512x32 tile (64 KB)
  const float* src = A + (size_t)img * 512 * 256 + cbase;
  for (int q = tid; q < 4096; q += TPB) {
    int r = q >> 3, seg = q & 7;
    async_copy_b128((char*)t + (size_t)q * 16,
                    (const void*)(src + (size_t)r * 256 + seg * 4));
  }
  wait_async_all();
  __syncthreads();

  const float p0 = py[0], p1 = py[1], p2 = py[2];
  const float u0 = uy[0], u1 = uy[1], u2 = uy[2];
  const float cc0 = cy[0], cc1 = cy[1], cc2 = cy[2];
  const float rr0 = ry[0], rr1 = ry[1], rr2 = ry[2];

  // in-place y lifting, evens = row 2j, odds = row 2j+1, M=256, 32 cols
  // step1: odd -= py * evens
  for (int it = tid; it < 8192; it += TPB) {
    int j = it >> 5, c = it & 31;
    int base = (2 * j) * 32 + c;
    float em = (j > 0)   ? t[base - 64] : 0.0f;
    float e0 = t[base];
    float ep = (j < 255) ? t[base + 64] : 0.0f;
    t[base + 32] -= p0 * em + p1 * e0 + p2 * ep;
  }
  __syncthreads();
  // step2: even += uy * odds (lvl 0 < 2 -> applied)
  for (int it = tid; it < 8192; it += TPB) {
    int j = it >> 5, c = it & 31;
    int base = (2 * j) * 32 + c;
    float om = (j > 0)   ? t[base - 32] : 0.0f;
    float o0 = t[base + 32];
    float op = (j < 255) ? t[base + 96] : 0.0f;
    t[base] += u0 * om + u1 * o0 + u2 * op;
  }
  __syncthreads();
  // step3: odd -= cy * evens
  for (int it = tid; it < 8192; it += TPB) {
    int j = it >> 5, c = it & 31;
    int base = (2 * j) * 32 + c;
    float em = (j > 0)   ? t[base - 64] : 0.0f;
    float e0 = t[base];
    float ep = (j < 255) ? t[base + 64] : 0.0f;
    t[base + 32] -= cc0 * em + cc1 * e0 + cc2 * ep;
  }
  __syncthreads();
  // step4: even += ry * odds
  for (int it = tid; it < 8192; it += TPB) {
    int j = it >> 5, c = it & 31;
    int base = (2 * j) * 32 + c;
    float om = (j > 0)   ? t[base - 32] : 0.0f;
    float o0 = t[base + 32];
    float op = (j < 255) ? t[base + 96] : 0.0f;
    t[base] += rr0 * om + rr1 * o0 + rr2 * op;
  }
  __syncthreads();

  // yo stats + write compacted ye
  float ss1 = 0.0f;
  unsigned* mh1 = &h1[((tid >> 5) & 3) * 256];
  for (int it = tid; it < 8192; it += TPB) {
    int j = it >> 5, c = it & 31;
    float v = t[(2 * j + 1) * 32 + c];
    float d = fmodf(v + 1.0f, 2.0f) - 1.0f;
    ss1 += d * d;
    hist_add(mh1, d);
    B[((size_t)img * 256 + j) * 256 + cbase + c] = t[(2 * j) * 32 + c];
  }
  atomicAdd(&ssh, ss1);
  __syncthreads();
  for (int b = tid; b < 256; b += TPB) {
    unsigned s = h1[b] + h1[256 + b] + h1[512 + b] + h1[768 + b];
    if (s) atomicAdd(&gh1[img * 256 + b], s);
  }
  if (tid == 0) atomicAdd(&gss[1], ssh);
}

// =====================================================================
// k3: levels 1..4 fully in LDS (256x256 = 256 KB tile), in-place with
// stride doubling. One workgroup per image, 1024 threads. All values in
// the tile are final coefficients at the end -> stats from LDS, no store.
// =====================================================================
__global__ __launch_bounds__(1024) void k3_deep_levels(
    const float* __restrict__ Bsrc,
    const float* __restrict__ px, const float* __restrict__ ux,
    const float* __restrict__ cx, const float* __restrict__ rx,
    const float* __restrict__ py, const float* __restrict__ uy,
    const float* __restrict__ cy, const float* __restrict__ ry,
    unsigned* __restrict__ gh1, float* __restrict__ gss)
{
  __shared__ __align__(16) float t[256 * 256];   // 256 KB (gfx1250: 320 KB LDS)
  __shared__ unsigned h1[4 * 256];
  __shared__ float ssh;
  const int tid = threadIdx.x;
  const int img = blockIdx.x;

  if (tid < 1024) h1[tid] = 0u;
  if (tid == 0) ssh = 0.0f;

  // async load the whole compacted image (256 KB)
  const char* src = (const char*)(Bsrc + (size_t)img * 65536);
  for (int q = tid; q < 16384; q += 1024)
    async_copy_b128((char*)t + (size_t)q * 16, src + (size_t)q * 16);
  wait_async_all();
  __syncthreads();

  const float xp0 = px[0], xp1 = px[1], xp2 = px[2];
  const float xu0 = ux[0], xu1 = ux[1], xu2 = ux[2];
  const float xc0 = cx[0], xc1 = cx[1], xc2 = cx[2];
  const float xr0 = rx[0], xr1 = rx[1], xr2 = rx[2];
  const float yp0 = py[0], yp1 = py[1], yp2 = py[2];
  const float yu0 = uy[0], yu1 = uy[1], yu2 = uy[2];
  const float yc0 = cy[0], yc1 = cy[1], yc2 = cy[2];
  const float yr0 = ry[0], yr1 = ry[1], yr2 = ry[2];

  for (int s = 0; s < 4; ++s) {            // lvl = s + 1
    const int st = 1 << s;
    const int M  = 128 >> s;               // half-size of current grid
    const int Mb = 7 - s;                  // log2(M)
    const int G  = 256 >> s;               // current grid size
    const int rowst = st * 256;
    const int nx = G * M;

    // ---- x-lift over all G rows of the strided grid ----
    for (int it = tid; it < nx; it += 1024) {       // odd -= px*evens
      int i = it >> Mb, j = it & (M - 1);
      int base = i * rowst + 2 * j * st;
      float em = (j > 0)     ? t[base - 2 * st] : 0.0f;
      float e0 = t[base];
      float ep = (j < M - 1) ? t[base + 2 * st] : 0.0f;
      t[base + st] -= xp0 * em + xp1 * e0 + xp2 * ep;
    }
    __syncthreads();
    for (int it = tid; it < nx; it += 1024) {       // even += ux*odds
      int i = it >> Mb, j = it & (M - 1);
      int base = i * rowst + 2 * j * st;
      float om = (j > 0)     ? t[base - st] : 0.0f;
      float o0 = t[base + st];
      float op = (j < M - 1) ? t[base + 3 * st] : 0.0f;
      t[base] += xu0 * om + xu1 * o0 + xu2 * op;
    }
    __syncthreads();
    for (int it = tid; it < nx; it += 1024) {       // odd -= cx*evens
      int i = it >> Mb, j = it & (M - 1);
      int base = i * rowst + 2 * j * st;
      float em = (j > 0)     ? t[base - 2 * st] : 0.0f;
      float e0 = t[base];
      float ep = (j < M - 1) ? t[base + 2 * st] : 0.0f;
      t[base + st] -= xc0 * em + xc1 * e0 + xc2 * ep;
    }
    __syncthreads();
    for (int it = tid; it < nx; it += 1024) {       // even += rx*odds
      int i = it >> Mb, j = it & (M - 1);
      int base = i * rowst + 2 * j * st;
      float om = (j > 0)     ? t[base - st] : 0.0f;
      float o0 = t[base + st];
      float op = (j < M - 1) ? t[base + 3 * st] : 0.0f;
      t[base] += xr0 * om + xr1 * o0 + xr2 * op;
    }
    __syncthreads();

    // ---- y-lift over even columns (xe) of the strided grid ----
    const int ny = M * M;
    for (int it = tid; it < ny; it += 1024) {       // odd -= py*evens
      int i = it >> Mb, j = it & (M - 1);
      int base = 2 * i * rowst + 2 * j * st;
      float em = (i > 0)     ? t[base - 2 * rowst] : 0.0f;
      float e0 = t[base];
      float ep = (i < M - 1) ? t[base + 2 * rowst] : 0.0f;
      t[base + rowst] -= yp0 * em + yp1 * e0 + yp2 * ep;
    }
    __syncthreads();
    if (s == 0) {                                   // uy only for lvl < 2
      for (int it = tid; it < ny; it += 1024) {     // even += uy*odds
        int i = it >> Mb, j = it & (M - 1);
        int base = 2 * i * rowst + 2 * j * st;
        float om = (i > 0)     ? t[base - rowst] : 0.0f;
        float o0 = t[base + rowst];
        float op = (i < M - 1) ? t[base + 3 * rowst] : 0.0f;
        t[base] += yu0 * om + yu1 * o0 + yu2 * op;
      }
      __syncthreads();
    }
    for (int it = tid; it < ny; it += 1024) {       // odd -= cy*evens
      int i = it >> Mb, j = it & (M - 1);
      int base = 2 * i * rowst + 2 * j * st;
      float em = (i > 0)     ? t[base - 2 * rowst] : 0.0f;
      float e0 = t[base];
      float ep = (i < M - 1) ? t[base + 2 * rowst] : 0.0f;
      t[base + rowst] -= yc0 * em + yc1 * e0 + yc2 * ep;
    }
    __syncthreads();
    for (int it = tid; it < ny; it += 1024) {       // even += ry*odds
      int i = it >> Mb, j = it & (M - 1);
      int base = 2 * i * rowst + 2 * j * st;
      float om = (i > 0)     ? t[base - rowst] : 0.0f;
      float o0 = t[base + rowst];
      float op = (i < M - 1) ? t[base + 3 * rowst] : 0.0f;
      t[base] += yr0 * om + yr1 * o0 + yr2 * op;
    }
    __syncthreads();
  }

  // ---- stats over the whole tile (every value is a final coefficient) ----
  float ss1 = 0.0f;
  unsigned* mh1 = &h1[((tid >> 5) & 3) * 256];
  for (int q = tid; q < 65536; q += 1024) {
    float d = fmodf(t[q] + 1.0f, 2.0f) - 1.0f;
    ss1 += d * d;
    hist_add(mh1, d);
  }
  atomicAdd(&ssh, ss1);
  __syncthreads();
  if (tid < 256) {
    unsigned sm = h1[tid] + h1[256 + tid] + h1[512 + tid] + h1[768 + tid];
    if (sm) atomicAdd(&gh1[img * 256 + tid], sm);
  }
  if (tid == 0) atomicAdd(&gss[1], ssh);
}

// =====================================================================
// k4: entropies + losses -> 4 scalars
// =====================================================================
__global__ __launch_bounds__(TPB) void k4_final(
    const unsigned* __restrict__ gh0, const unsigned* __restrict__ gh1,
    const float* __restrict__ gss, float* __restrict__ out)
{
  __shared__ float ra[TPB], rb_[TPB];
  const int tid = threadIdx.x;
  float e0 = 0.0f, e1 = 0.0f;
  const float inv_res = 1.0f / 262144.0f;
  for (int i = tid; i < HIST_N; i += TPB) {
    unsigned a = gh0[i];
    if (a) { float p = (float)a * inv_res; e0 -= p * log2f(p); }
    unsigned b = gh1[i];
    if (b) { float p = (float)b * inv_res; e1 -= p * log2f(p); }
  }
  ra[tid] = e0; rb_[tid] = e1;
  __syncthreads();
  for (int s = TPB / 2; s > 0; s >>= 1) {
    if (tid < s) { ra[tid] += ra[tid + s]; rb_[tid] += rb_[tid + s]; }
    __syncthreads();
  }
  if (tid == 0) {
    const float denom = 96.0f * 262144.0f;
    out[0] = 255.0f * sqrtf(gss[1] / denom);  // loss1 (deltas)
    out[1] = 255.0f * sqrtf(gss[0] / denom);  // loss0 (input)
    out[2] = ra[0] / (8.0f * 96.0f);          // invCR0
    out[3] = rb_[0] / (8.0f * 96.0f);         // invCR1
  }
}

// =====================================================================
extern "C" void kernel_launch(void* const* d_in, const int* in_sizes, int n_in,
                              void* d_out, int out_size, void* d_ws, size_t ws_size,
                              hipStream_t stream)
{
  (void)in_sizes; (void)n_in; (void)out_size; (void)ws_size;
  const float* x  = (const float*)d_in[0];
  const float* px = (const float*)d_in[1];
  const float* ux = (const float*)d_in[2];
  const float* cx = (const float*)d_in[3];
  const float* rx = (const float*)d_in[4];
  const float* py = (const float*)d_in[5];
  const float* uy = (const float*)d_in[6];
  const float* cy = (const float*)d_in[7];
  const float* ry = (const float*)d_in[8];

  float*    A   = (float*)d_ws;
  float*    Bb  = A + A_ELEMS;
  unsigned* gh0 = (unsigned*)(Bb + B_ELEMS);
  unsigned* gh1 = gh0 + HIST_N;
  float*    gss = (float*)(gh1 + HIST_N);
  float*    out = (float*)d_out;

  k0_init<<<(HIST_N + TPB - 1) / TPB, TPB, 0, stream>>>(gh0, gh1, gss);
  k1_xlift_stats<<<IMGS * 64, TPB, 0, stream>>>(x, px, ux, cx, rx, A, gh0, gh1, gss);
  k2_ylift_stats<<<IMGS * 8, TPB, 0, stream>>>(A, py, uy, cy, ry, Bb, gh1, gss);
  k3_deep_levels<<<IMGS, 1024, 0, stream>>>(Bb, px, ux, cx, rx, py, uy, cy, ry, gh1, gss);
  k4_final<<<1, TPB, 0, stream>>>(gh0, gh1, gss, out);
}